// GINEEncoder_52106543235221
// MI455X (gfx1250) — compile-verified
//
#include <hip/hip_runtime.h>

#define HID 128
#define LDS_PITCH 132   // 128 + 4 floats pad -> lane stride 132 % 64 banks = 4, conflict-free

typedef __attribute__((ext_vector_type(2))) float v2f;
typedef __attribute__((ext_vector_type(8))) float v8f;

// ---------------------------------------------------------------------------
// out[m][n] = relu( sum_k A[m][k] * W[k][n] + bias[n] )
// One workgroup = 8 waves = 16 output rows x 128 cols. Each wave owns a 16x16
// tile (colBase = wave*16) and runs 32 x V_WMMA_F32_16X16X4_F32 over K=128.
// A tile (16x128) staged in LDS once, reused by all 8 waves. W stays in L2.
// ---------------------------------------------------------------------------
__global__ __launch_bounds__(256) void gemm_bias_relu_kernel(
    const float* __restrict__ A, const float* __restrict__ W,
    const float* __restrict__ bias, float* __restrict__ out, int nRows)
{
    __shared__ float As[16 * LDS_PITCH];

    const int rowBase = blockIdx.x * 16;
    const int t = threadIdx.x;

    // Cooperative load of the 16x128 f32 A tile (512 float4 / 256 threads).
    for (int i = t; i < 16 * 32; i += 256) {
        const int r  = i >> 5;          // row 0..15
        const int c4 = (i & 31) << 2;   // col 0,4,...,124
        int gr = rowBase + r;
        if (gr >= nRows) gr = nRows - 1;            // clamp (exact-fit anyway)
        const float4 v = *reinterpret_cast<const float4*>(A + (size_t)gr * HID + c4);
        *reinterpret_cast<float4*>(&As[r * LDS_PITCH + c4]) = v;  // 528B row pitch: 16B aligned
    }
    __syncthreads();

    const int wave    = t >> 5;         // 0..7
    const int lane    = t & 31;
    const int lh      = lane >> 4;      // lane half: 0/1
    const int ll      = lane & 15;
    const int colBase = wave * 16;

    // A fragment source: lane ll = row M, K = 4*kk + 2*lh + {0,1}
    const float* Ar = &As[ll * LDS_PITCH + 2 * lh];
    // B fragment source: lane ll = col N, VGPR{0,1} = K rows {4kk+2lh, 4kk+2lh+1}
    const float* Wc = W + (size_t)(2 * lh) * HID + colBase + ll;

    v8f c = {};
#pragma unroll
    for (int kk = 0; kk < 32; ++kk) {
        const int k4 = kk * 4;
        v2f a, b;
        a.x = Ar[k4 + 0];
        a.y = Ar[k4 + 1];
        b.x = Wc[(size_t)k4 * HID];
        b.y = Wc[(size_t)(k4 + 1) * HID];
        // D = A(16x4) * B(4x16) + C   -> v_wmma_f32_16x16x4_f32
        c = __builtin_amdgcn_wmma_f32_16x16x4_f32(
                /*neg_a=*/false, a, /*neg_b=*/false, b,
                /*c_mod=*/(short)0, c, /*reuse_a=*/false, /*reuse_b=*/false);
    }

    // C/D layout: VGPR r -> (M = r + 8*lh, N = ll). Fuse bias + ReLU on store.
    const float bv = bias[colBase + ll];
    const int   row0 = rowBase + lh * 8;
    float* o = out + colBase + ll;
#pragma unroll
    for (int r = 0; r < 8; ++r) {
        float v = c[r] + bv;
        v = v > 0.0f ? v : 0.0f;
        o[(size_t)(row0 + r) * HID] = v;
    }
}

// ---------------------------------------------------------------------------
// One wave per edge: lane owns float4 of the 128-dim feature.
// msg = relu(h[src] + emb[etype]) ; agg[dst] += msg  (global f32 atomics)
// ---------------------------------------------------------------------------
__global__ __launch_bounds__(256) void edge_scatter_kernel(
    const float* __restrict__ h, const int* __restrict__ src,
    const int* __restrict__ dst, const int* __restrict__ etype,
    const float* __restrict__ emb, float* __restrict__ agg, int nEdges)
{
    const int w = (blockIdx.x * 256 + threadIdx.x) >> 5;   // edge id
    if (w >= nEdges) return;
    const int lane = threadIdx.x & 31;
    const int c4 = lane << 2;

    const int s  = src[w];
    const int d  = dst[w];
    const int ty = etype[w];

    const float4 hv = *reinterpret_cast<const float4*>(h   + (size_t)s  * HID + c4);
    const float4 ev = *reinterpret_cast<const float4*>(emb + (size_t)ty * HID + c4);

    float m0 = hv.x + ev.x; m0 = m0 > 0.0f ? m0 : 0.0f;
    float m1 = hv.y + ev.y; m1 = m1 > 0.0f ? m1 : 0.0f;
    float m2 = hv.z + ev.z; m2 = m2 > 0.0f ? m2 : 0.0f;
    float m3 = hv.w + ev.w; m3 = m3 > 0.0f ? m3 : 0.0f;

    float* a = agg + (size_t)d * HID + c4;
    atomicAdd(a + 0, m0);
    atomicAdd(a + 1, m1);
    atomicAdd(a + 2, m2);
    atomicAdd(a + 3, m3);
}

// Zero a float buffer (float4 granularity; n4 = #float4).
__global__ __launch_bounds__(256) void fill_zero_kernel(float4* __restrict__ p, long n4)
{
    const long i = (long)blockIdx.x * 256 + threadIdx.x;
    if (i < n4) p[i] = make_float4(0.f, 0.f, 0.f, 0.f);
}

// In-place: agg_u = (1 + *eps) * h + agg_u
__global__ __launch_bounds__(256) void combine_kernel(
    const float4* __restrict__ h, float4* __restrict__ agg_u,
    const float* __restrict__ epsPtr, long n4)
{
    const long i = (long)blockIdx.x * 256 + threadIdx.x;
    if (i >= n4) return;
    const float s = 1.0f + *epsPtr;
    const float4 hv = h[i];
    float4 av = agg_u[i];
    av.x = fmaf(s, hv.x, av.x);
    av.y = fmaf(s, hv.y, av.y);
    av.z = fmaf(s, hv.z, av.z);
    av.w = fmaf(s, hv.w, av.w);
    agg_u[i] = av;
}

extern "C" void kernel_launch(void* const* d_in, const int* in_sizes, int n_in,
                              void* d_out, int out_size, void* d_ws, size_t ws_size,
                              hipStream_t stream)
{
    const float* x    = (const float*)d_in[0];
    const int*   ei   = (const int*)  d_in[1];   // [2, E]
    const int*   eatt = (const int*)  d_in[2];   // [E]
    const float* Wx   = (const float*)d_in[3];
    const float* bx   = (const float*)d_in[4];
    const float* emb  = (const float*)d_in[5];
    const float* eps1 = (const float*)d_in[6];
    const float* eps2 = (const float*)d_in[7];
    const float* W1a = (const float*)d_in[8];   const float* b1a = (const float*)d_in[9];
    const float* W1b = (const float*)d_in[10];  const float* b1b = (const float*)d_in[11];
    const float* W1c = (const float*)d_in[12];  const float* b1c = (const float*)d_in[13];
    const float* W2a = (const float*)d_in[14];  const float* b2a = (const float*)d_in[15];
    const float* W2b = (const float*)d_in[16];  const float* b2b = (const float*)d_in[17];
    const float* W2c = (const float*)d_in[18];  const float* b2c = (const float*)d_in[19];

    const int nNodes = in_sizes[0] / HID;       // 100000
    const int nEdges = in_sizes[2];             // 1600000
    const int* srcI = ei;
    const int* dstI = ei + nEdges;

    const size_t nbuf = (size_t)nNodes * HID * sizeof(float);   // 51.2 MB
    float* B0 = (float*)d_ws;                                   // ping-pong node bufs
    float* B1 = (float*)((char*)d_ws + nbuf);
    float* B2 = (float*)((char*)d_ws + 2 * nbuf);
    float* out = (float*)d_out;

    const long n4 = (long)nNodes * HID / 4;
    const int  ewBlocks   = (int)((n4 + 255) / 256);
    const int  gemmBlocks = (nNodes + 15) / 16;                 // 6250, exact
    const int  edgeBlocks = (int)(((long)nEdges * 32 + 255) / 256);

    // h0 = relu(x @ Wx + bx)
    gemm_bias_relu_kernel<<<gemmBlocks, 256, 0, stream>>>(x, Wx, bx, B0, nNodes);

    // ---- GINE layer 1 ----
    fill_zero_kernel<<<ewBlocks, 256, 0, stream>>>((float4*)B1, n4);
    edge_scatter_kernel<<<edgeBlocks, 256, 0, stream>>>(B0, srcI, dstI, eatt, emb, B1, nEdges);
    combine_kernel<<<ewBlocks, 256, 0, stream>>>((const float4*)B0, (float4*)B1, eps1, n4);
    gemm_bias_relu_kernel<<<gemmBlocks, 256, 0, stream>>>(B1, W1a, b1a, B2, nNodes);
    gemm_bias_relu_kernel<<<gemmBlocks, 256, 0, stream>>>(B2, W1b, b1b, B1, nNodes);
    gemm_bias_relu_kernel<<<gemmBlocks, 256, 0, stream>>>(B1, W1c, b1c, B0, nNodes); // h1

    // ---- GINE layer 2 ----
    fill_zero_kernel<<<ewBlocks, 256, 0, stream>>>((float4*)B2, n4);
    edge_scatter_kernel<<<edgeBlocks, 256, 0, stream>>>(B0, srcI, dstI, eatt, emb, B2, nEdges);
    combine_kernel<<<ewBlocks, 256, 0, stream>>>((const float4*)B0, (float4*)B2, eps2, n4);
    gemm_bias_relu_kernel<<<gemmBlocks, 256, 0, stream>>>(B2, W2a, b2a, B1, nNodes);
    gemm_bias_relu_kernel<<<gemmBlocks, 256, 0, stream>>>(B1, W2b, b2b, B2, nNodes);
    gemm_bias_relu_kernel<<<gemmBlocks, 256, 0, stream>>>(B2, W2c, b2c, out, nNodes); // final relu'd h
}